// KGpriorPredictor_39625368273220
// MI455X (gfx1250) — compile-verified
//
#include <hip/hip_runtime.h>
#include <hip/hip_bf16.h>

typedef __attribute__((ext_vector_type(4))) float v4f;

#define NOC 151u   // object classes
#define NRC 51u    // relation classes

// ---------------------------------------------------------------------------
// obj_dists: exact one-hot expansion (softmax of +-1000 logits underflows to
// one-hot in fp32). One float4 per thread, non-temporal b128 stores.
// ---------------------------------------------------------------------------
__global__ __launch_bounds__(256) void obj_onehot_v4(const int* __restrict__ labels,
                                                     float* __restrict__ out,
                                                     unsigned n4, unsigned nrows) {
    unsigned gid = blockIdx.x * blockDim.x + threadIdx.x;
    if (gid >= n4) return;
    unsigned e0  = gid * 4u;
    unsigned row = e0 / NOC;
    unsigned col = e0 - row * NOC;
    unsigned lab0 = (unsigned)labels[row];
    unsigned row1 = (row + 1u < nrows) ? row + 1u : row;
    // crossing into next row only if col >= 148
    unsigned lab1 = (col > NOC - 4u) ? (unsigned)labels[row1] : lab0;
    v4f v;
#pragma unroll
    for (int j = 0; j < 4; ++j) {
        unsigned c = col + (unsigned)j;
        unsigned lab = lab0;
        if (c >= NOC) { c -= NOC; lab = lab1; }
        v[j] = (c == lab) ? 1.0f : 0.0f;
    }
    __builtin_nontemporal_store(v, (v4f*)out + gid);
}

// ---------------------------------------------------------------------------
// rel_dists: row gather from prior_table. One float4 of the flat output per
// thread: e = 4*gid, p = e/51, r = e%51. A float4 spans at most 2 pairs.
// Table reads hit L2/WGP$ (4.65 MB table, NT stores keep it resident).
// ---------------------------------------------------------------------------
__global__ __launch_bounds__(256) void rel_gather_v4(const int* __restrict__ labels,
                                                     const int2* __restrict__ pairs,
                                                     const float* __restrict__ table,
                                                     float* __restrict__ out,
                                                     unsigned n4) {
    unsigned gid = blockIdx.x * blockDim.x + threadIdx.x;
    if (gid >= n4) return;
    unsigned e0 = gid * 4u;
    unsigned p  = e0 / NRC;
    unsigned r  = e0 - p * NRC;

    // Pull the pair-index stream ahead (gfx1250 global_prefetch_b8;
    // speculative prefetch is silently dropped if the address is unmapped).
    __builtin_prefetch(&pairs[p + 2048u], 0, 1);

    int2 pr0 = pairs[p];
    const float* row0 =
        table + ((unsigned)labels[pr0.x] * NOC + (unsigned)labels[pr0.y]) * NRC;

    v4f v;
    if (r <= NRC - 4u) {           // all four elements inside this pair's row
        v.x = row0[r];
        v.y = row0[r + 1u];
        v.z = row0[r + 2u];
        v.w = row0[r + 3u];
    } else {                       // straddles into pair p+1 (r in {48,49,50})
        int2 pr1 = pairs[p + 1u];
        const float* row1 =
            table + ((unsigned)labels[pr1.x] * NOC + (unsigned)labels[pr1.y]) * NRC;
#pragma unroll
        for (int j = 0; j < 4; ++j) {
            unsigned rr = r + (unsigned)j;
            v[j] = (rr < NRC) ? row0[rr] : row1[rr - NRC];
        }
    }
    __builtin_nontemporal_store(v, (v4f*)out + gid);
}

// ---------------------------------------------------------------------------
// Scalar fallback kernels (only used if shapes break the float4 invariants).
// ---------------------------------------------------------------------------
__global__ __launch_bounds__(256) void obj_onehot_scalar(const int* __restrict__ labels,
                                                         float* __restrict__ out,
                                                         unsigned n) {
    unsigned e = blockIdx.x * blockDim.x + threadIdx.x;
    if (e >= n) return;
    unsigned row = e / NOC;
    unsigned col = e - row * NOC;
    out[e] = (col == (unsigned)labels[row]) ? 1.0f : 0.0f;
}

__global__ __launch_bounds__(256) void rel_gather_scalar(const int* __restrict__ labels,
                                                         const int2* __restrict__ pairs,
                                                         const float* __restrict__ table,
                                                         float* __restrict__ out,
                                                         unsigned n) {
    unsigned e = blockIdx.x * blockDim.x + threadIdx.x;
    if (e >= n) return;
    unsigned p = e / NRC;
    unsigned r = e - p * NRC;
    int2 pr = pairs[p];
    out[e] = table[((unsigned)labels[pr.x] * NOC + (unsigned)labels[pr.y]) * NRC + r];
}

extern "C" void kernel_launch(void* const* d_in, const int* in_sizes, int n_in,
                              void* d_out, int out_size, void* d_ws, size_t ws_size,
                              hipStream_t stream) {
    (void)n_in; (void)d_ws; (void)ws_size; (void)out_size;

    const int*   labels = (const int*)d_in[0];
    const int2*  pairs  = (const int2*)d_in[1];
    const float* table  = (const float*)d_in[2];
    float*       out    = (float*)d_out;

    const unsigned n_obj     = (unsigned)in_sizes[0];          // 16384
    const unsigned n_pairs   = (unsigned)in_sizes[1] / 2u;     // 2,000,000
    const unsigned obj_elems = n_obj * NOC;                    // 2,473,984
    const unsigned rel_elems = n_pairs * NRC;                  // 102,000,000

    // float4 fast path requires both sections 4-float divisible (which also
    // makes the rel section 16B-aligned relative to d_out).
    const bool fast = ((obj_elems & 3u) == 0u) && ((rel_elems & 3u) == 0u);

    if (fast) {
        const unsigned obj_n4 = obj_elems / 4u;                // 618,496
        const unsigned rel_n4 = rel_elems / 4u;                // 25,500,000
        obj_onehot_v4<<<(obj_n4 + 255u) / 256u, 256, 0, stream>>>(
            labels, out, obj_n4, n_obj);
        rel_gather_v4<<<(rel_n4 + 255u) / 256u, 256, 0, stream>>>(
            labels, pairs, table, out + obj_elems, rel_n4);
    } else {
        obj_onehot_scalar<<<(obj_elems + 255u) / 256u, 256, 0, stream>>>(
            labels, out, obj_elems);
        rel_gather_scalar<<<(rel_elems + 255u) / 256u, 256, 0, stream>>>(
            labels, pairs, table, out + obj_elems, rel_elems);
    }
}